// runLayer_27247272526218
// MI455X (gfx1250) — compile-verified
//
#include <hip/hip_runtime.h>
#include <math.h>

typedef __attribute__((ext_vector_type(2))) float v2f;
typedef __attribute__((ext_vector_type(8))) float v8f;

#define C_DIM 512
#define N_DIM 784          // 28*28 = 49 * 16
#define CT_TILES (N_DIM / 16)   // 49 column tiles
#define EPS_F 1e-5f
#define SUB_SCALE 0.6f

__global__ __launch_bounds__(1024) void spectral_sub_kernel(
    const float* __restrict__ A_in,   // [B, 512, 784]
    const float* __restrict__ v0_in,  // [B, 784, 1]
    float* __restrict__ out)          // [B, 512, 784]
{
    __shared__ float s_v[N_DIM];
    __shared__ float s_u[C_DIM];
    __shared__ float s_red;

    const int b      = blockIdx.x;
    const int t      = threadIdx.x;
    const int lane   = t & 31;
    const int wave   = t >> 5;           // 0..31
    const float* Ab  = A_in + (size_t)b * C_DIM * N_DIM;
    float*       Ob  = out  + (size_t)b * C_DIM * N_DIM;

    // ---- load v0 into LDS ----
    if (t < N_DIM) s_v[t] = v0_in[(size_t)b * N_DIM + t];
    __syncthreads();

    // ---- 2 power iterations: v <- A^T (A v) ----
    for (int iter = 0; iter < 2; ++iter) {
        // u = A v  (wave-per-row, shuffle reduction; coalesced row reads)
        for (int c = wave; c < C_DIM; c += 32) {
            const float* row = Ab + (size_t)c * N_DIM;
            float acc = 0.0f;
            for (int n = lane; n < N_DIM; n += 32)
                acc = fmaf(row[n], s_v[n], acc);
            #pragma unroll
            for (int off = 16; off > 0; off >>= 1)
                acc += __shfl_down(acc, off, 32);
            if (lane == 0) s_u[c] = acc;
        }
        __syncthreads();

        // v = A^T u  (thread-per-column; each c-step is a coalesced row sweep,
        // s_u[c] is a uniform LDS broadcast)
        if (t < N_DIM) {
            float acc = 0.0f;
            #pragma unroll 8
            for (int c = 0; c < C_DIM; ++c)
                acc = fmaf(Ab[(size_t)c * N_DIM + t], s_u[c], acc);
            s_v[t] = acc;
        }
        __syncthreads();
    }

    // ---- normalize: v /= sqrt(||v||^2 + eps) ----
    if (t == 0) s_red = 0.0f;
    __syncthreads();
    {
        float x  = (t < N_DIM) ? s_v[t] : 0.0f;
        float sq = x * x;
        #pragma unroll
        for (int off = 16; off > 0; off >>= 1)
            sq += __shfl_down(sq, off, 32);
        if (lane == 0) atomicAdd(&s_red, sq);   // ds_add_f32
    }
    __syncthreads();
    const float inv = rsqrtf(s_red + EPS_F);
    if (t < N_DIM) s_v[t] *= inv;
    __syncthreads();

    // ---- final u = A v, pre-scaled by -0.6 ----
    for (int c = wave; c < C_DIM; c += 32) {
        const float* row = Ab + (size_t)c * N_DIM;
        float acc = 0.0f;
        for (int n = lane; n < N_DIM; n += 32)
            acc = fmaf(row[n], s_v[n], acc);
        #pragma unroll
        for (int off = 16; off > 0; off >>= 1)
            acc += __shfl_down(acc, off, 32);
        if (lane == 0) s_u[c] = -SUB_SCALE * acc;
    }
    __syncthreads();

    // ---- out = A + (-0.6 u) v^T  via V_WMMA_F32_16X16X4_F32 (K=1 of 4) ----
    // Static wave -> row-tile map: 32 waves == 32 row tiles of 512 rows.
    // A-operand 16x4 f32 (2 VGPRs): VGPR0 = {K=0 | K=2} by lane half.
    //   Only K=0 nonzero: lanes 0-15 carry -0.6*u[m0+lane]; loop-invariant.
    // B-operand 4x16 mirrors: lanes 0-15 of .x carry v[n0+lane].
    // C/D 16x16: VGPR i, lane l -> row m0 + i + 8*(l>=16), col n0 + (l&15).
    {
        const int halfSel = lane >> 4;      // 0: lanes 0-15, 1: lanes 16-31
        const int lane16  = lane & 15;
        const int m0      = wave * 16;

        // Unconditional LDS load + select (v_cndmask, no EXEC branching).
        const float uval = s_u[m0 + lane16];
        v2f av;
        av.x = halfSel ? 0.0f : uval;       // K=0 slot only
        av.y = 0.0f;                        // K=1 / K=3 slots

        const size_t rowoff = (size_t)(m0 + 8 * halfSel) * N_DIM + lane16;

        for (int ct = 0; ct < CT_TILES; ++ct) {
            const int n0 = ct * 16;

            const float vval = s_v[n0 + lane16];
            v2f bv;
            bv.x = halfSel ? 0.0f : vval;   // K=0 slot only
            bv.y = 0.0f;

            const size_t cbase = rowoff + (size_t)n0;
            v8f cc;
            #pragma unroll
            for (int i = 0; i < 8; ++i)
                cc[i] = Ab[cbase + (size_t)i * N_DIM];

            v8f d = __builtin_amdgcn_wmma_f32_16x16x4_f32(
                /*neg_a=*/false, av, /*neg_b=*/false, bv,
                /*c_mod=*/(short)0, cc, /*reuse_a=*/false, /*reuse_b=*/false);

            // Non-temporal stores: keep A resident in the 192MB L2 for the
            // remaining passes of concurrently-running batches.
            float* ob = Ob + cbase;
            #pragma unroll
            for (int i = 0; i < 8; ++i)
                __builtin_nontemporal_store(d[i], &ob[(size_t)i * N_DIM]);
        }
    }
}

extern "C" void kernel_launch(void* const* d_in, const int* in_sizes, int n_in,
                              void* d_out, int out_size, void* d_ws, size_t ws_size,
                              hipStream_t stream) {
    (void)in_sizes; (void)n_in; (void)d_ws; (void)ws_size; (void)out_size;
    const float* A  = (const float*)d_in[0];   // input  [64, 512, 28, 28] f32
    const float* v0 = (const float*)d_in[1];   // v0     [64, 784, 1]      f32
    float* out = (float*)d_out;                // out    [64, 512, 784]    f32

    const int B = 64;
    spectral_sub_kernel<<<B, 1024, 0, stream>>>(A, v0, out);
}